// ChamferLoss3D_27960237097114
// MI455X (gfx1250) — compile-verified
//
#include <hip/hip_runtime.h>
#include <hip/hip_bf16.h>
#include <stdint.h>

typedef float v2f __attribute__((ext_vector_type(2)));
typedef float v8f __attribute__((ext_vector_type(8)));

#define BB 8
#define MM 4096
#define NN 4096
#define EPSF 1e-8f
#define THREADS 256
#define MTILE 128                         // 8 waves * 16 rows each
// LDS: B-operand pairs (2*NN float2 = 64KB) + gsq (16KB) + backward mins (16KB)
#define LDS_BYTES (2 * NN * 8 + NN * 4 + NN * 4)

// ---------------------------------------------------------------------------
// Init: zero scalar output, set backward per-(b,n) min array to +inf bits.
// ---------------------------------------------------------------------------
__global__ void chamfer_init_kernel(unsigned* bwd, float* out) {
    int i = blockIdx.x * blockDim.x + threadIdx.x;
    if (i < BB * NN) bwd[i] = 0x7F800000u;   // +inf (float bits)
    if (i == 0) out[0] = 0.0f;
}

// ---------------------------------------------------------------------------
// Main: one block = 128 predict rows of one batch; sweeps all N tiles.
// v_wmma_f32_16x16x4_f32 produces 16x16 tiles of p.g dot products (K=3, zero-pad).
// ---------------------------------------------------------------------------
__global__ void __launch_bounds__(THREADS)
chamfer_main_kernel(const float* __restrict__ p,   // [B,3,M]
                    const float* __restrict__ g,   // [B,3,N]
                    float* __restrict__ out,       // scalar accumulator
                    unsigned* __restrict__ bwd)    // [B,N] float-bit min dists
{
    extern __shared__ char smem[];
    v2f*      sB2  = (v2f*)smem;                             // [2*NN]: (x,y) | (z,0)
    float*    sGsq = (float*)(smem + 2 * NN * 8);            // [NN]
    unsigned* sbwd = (unsigned*)(smem + 2 * NN * 8 + NN * 4);// [NN]

    const int b    = blockIdx.y;
    const int tid  = threadIdx.x;
    const int wave = tid >> 5;
    const int lane = tid & 31;
    const int half = lane >> 4;     // 0: lanes 0-15 (K=0,1), 1: lanes 16-31 (K=2,3)
    const int l16  = lane & 15;

    // Stage gt cloud: pre-packed WMMA-B operand pairs + ||g||^2 + bwd-min init.
    const float* gb = g + (size_t)b * 3 * NN;
    for (int i = tid; i < NN; i += THREADS) {
        const float x = gb[i], y = gb[NN + i], z = gb[2 * NN + i];
        v2f lo; lo.x = x; lo.y = y;
        v2f hi; hi.x = z; hi.y = 0.0f;
        sB2[i]      = lo;
        sB2[NN + i] = hi;
        sGsq[i]     = x * x + y * y + z * z;
        sbwd[i]     = 0x7F800000u;
    }
    __syncthreads();

    const float* pb    = p + (size_t)b * 3 * MM;
    const int    mbase = blockIdx.x * MTILE + wave * 16;

    // A-matrix (16x4 f32, ISA layout): M = l16; low half holds K=0,1 (x,y),
    // high half holds K=2,3 (z, zero-pad).
    const int am = mbase + l16;
    v2f A;
    A.x = half ? pb[2 * MM + am] : pb[am];
    A.y = half ? 0.0f            : pb[MM + am];

    // ||p||^2 for the 8 C-layout rows of this lane (VGPR v -> M = v + 8*half).
    float psq[8];
#pragma unroll
    for (int v = 0; v < 8; ++v) {
        const int r = mbase + half * 8 + v;
        const float x = pb[r], y = pb[MM + r], z = pb[2 * MM + r];
        psq[v] = x * x + y * y + z * z;
    }

    float bestF[8];
#pragma unroll
    for (int v = 0; v < 8; ++v) bestF[v] = 3.0e38f;

    const int bsel = half * NN;   // B-operand table select for this half

    // Sweep all gt points in 16-wide tiles: no shuffles, no divergence.
    for (int n0 = 0; n0 < NN; n0 += 16) {
        const int gn  = n0 + l16;
        const v2f Bv  = sB2[bsel + gn];   // single ds_load_b64
        const float gsq = sGsq[gn];       // ds_load_b32

        v8f C = {};
        C = __builtin_amdgcn_wmma_f32_16x16x4_f32(
                /*neg_a=*/false, A, /*neg_b=*/false, Bv,
                /*c_mod=*/(short)0, C, /*reuse_a=*/false, /*reuse_b=*/false);

        float fb = 3.0e38f;               // backward partial over this lane's rows
#pragma unroll
        for (int v = 0; v < 8; ++v) {
            const float t = fmaf(-2.0f, C[v], psq[v]);  // psq - 2*dot
            bestF[v] = fminf(bestF[v], t + gsq);        // forward running min
            fb       = fminf(fb, t);                    // backward row min
        }
        // Both halves contribute their 8-row partial; no-return LDS atomic min.
        atomicMin(&sbwd[gn], __float_as_uint(fmaxf(fb + gsq, 0.0f)));
    }

    // Forward: reduce each row's min across the 16 lanes of its half.
#pragma unroll
    for (int v = 0; v < 8; ++v) {
        float d = bestF[v];
        d = fminf(d, __shfl_xor(d, 1, 32));
        d = fminf(d, __shfl_xor(d, 2, 32));
        d = fminf(d, __shfl_xor(d, 4, 32));
        d = fminf(d, __shfl_xor(d, 8, 32));
        bestF[v] = d;
    }
    if (l16 == 0) {   // lane 0 (rows 0-7) and lane 16 (rows 8-15)
        float s = 0.0f;
#pragma unroll
        for (int v = 0; v < 8; ++v) s += sqrtf(fmaxf(bestF[v], 0.0f) + EPSF);
        atomicAdd(out, s * (1.0f / (float)(BB * MM)));
    }

    __syncthreads();
    // Flush block-local backward mins to the global per-(b,n) array.
    unsigned* gB = bwd + (size_t)b * NN;
    for (int i = tid; i < NN; i += THREADS) atomicMin(&gB[i], sbwd[i]);
}

// ---------------------------------------------------------------------------
// Finalize: backward loss = mean over (b,n) of sqrt(minDist + eps).
// ---------------------------------------------------------------------------
__global__ void chamfer_bwd_finalize_kernel(const unsigned* __restrict__ bwd,
                                            float* __restrict__ out) {
    const int i = blockIdx.x * blockDim.x + threadIdx.x;
    float v = 0.0f;
    if (i < BB * NN) {
        const float d = __uint_as_float(bwd[i]);
        v = sqrtf(d + EPSF);
    }
#pragma unroll
    for (int m = 1; m < 32; m <<= 1) v += __shfl_xor(v, m, 32);

    __shared__ float ws[8];
    const int wave = threadIdx.x >> 5, lane = threadIdx.x & 31;
    if (lane == 0) ws[wave] = v;
    __syncthreads();
    if (threadIdx.x == 0) {
        float s = 0.0f;
#pragma unroll
        for (int w = 0; w < 8; ++w) s += ws[w];
        atomicAdd(out, s * (1.0f / (float)(BB * NN)));
    }
}

// ---------------------------------------------------------------------------
extern "C" void kernel_launch(void* const* d_in, const int* in_sizes, int n_in,
                              void* d_out, int out_size, void* d_ws, size_t ws_size,
                              hipStream_t stream) {
    const float* p   = (const float*)d_in[0];   // predict_pc [B,3,M]
    const float* g   = (const float*)d_in[1];   // gt_pc      [B,3,N]
    float*       out = (float*)d_out;           // scalar loss
    unsigned*    bwd = (unsigned*)d_ws;         // [B*N] float-bit mins (128KB)

    const int total = BB * NN;
    chamfer_init_kernel<<<(total + 255) / 256, 256, 0, stream>>>(bwd, out);

    dim3 grid(MM / MTILE, BB);
    chamfer_main_kernel<<<grid, THREADS, LDS_BYTES, stream>>>(p, g, out, bwd);

    chamfer_bwd_finalize_kernel<<<(total + 255) / 256, 256, 0, stream>>>(bwd, out);
}